// TempleGate_7808250544376
// MI455X (gfx1250) — compile-verified
//
#include <hip/hip_runtime.h>
#include <hip/hip_bf16.h>
#include <stdint.h>

#define CHAOS_R 3.5699456f
#define N_STEPS 10
#define BATCH   256
#define TLEN    8192
#define TWO_T   16384
#define TILE    256            // te elements per block
#define TILE_DW (TILE * 5)     // 1280 dwords staged per block

// jnp.mod(r, 10) == r - 10*floor(r/10)  (Python-style, sign of divisor)
__device__ __forceinline__ float mod10(float r) {
    float q = floorf(r * 0.1f);
    float m = fmaf(-10.0f, q, r);
    if (m < 0.0f)    m += 10.0f;   // rounding guards
    if (m >= 10.0f)  m -= 10.0f;
    return m;
}

// ---------------------------------------------------------------------------
// Prep kernel: scalar seed/chaos/mode + folded weight vectors.
//   resonance = breath . (temple @ vec) == w . vec
//   w_main[d]   = sum_r temple[r][d] * breath[r]     (r = 0..3)
//   w_mirror[d] = sum_r temple[r][d] * breath[r+1]
// ws layout (floats): [0..4]=w_main  [5..9]=w_mirror  [10 as int]=mode
// ---------------------------------------------------------------------------
__global__ void TempleGate_prep(const float* __restrict__ x,
                                const float* __restrict__ temple,
                                const float* __restrict__ breath,
                                float* __restrict__ wsf,
                                float* __restrict__ out,
                                int out_size) {
    float mean = 0.0f;
    #pragma unroll
    for (int d = 0; d < 5; ++d) mean += x[d];
    mean *= 0.2f;
    float seed = mean - floorf(mean);          // mod(mean, 1.0), exact
    float chaos = CHAOS_R * seed * (1.0f - seed);
    int mode = (chaos > 0.7f) ? 0 : ((chaos > 0.4f) ? 1 : 2);

    #pragma unroll
    for (int d = 0; d < 5; ++d) {
        float wm = 0.0f, wx = 0.0f;
        #pragma unroll
        for (int r = 0; r < 4; ++r) {
            wm += temple[r * 5 + d] * breath[r];
            wx += temple[r * 5 + d] * breath[r + 1];
        }
        wsf[d]     = wm;
        wsf[5 + d] = wx;
    }
    ((int*)wsf)[10] = mode;
    out[out_size - 1] = (float)mode;   // mode_code output
}

// ---------------------------------------------------------------------------
// Main kernel: one block = one batch row b and 256 consecutive te values.
// Source region (contiguous even for flip mode) is staged in via gfx1250
// async global->LDS loads; extended_x output is staged out via async
// LDS->global stores for fully coalesced contiguous dword traffic.
// ---------------------------------------------------------------------------
__global__ void __launch_bounds__(256)
TempleGate_main(const float* __restrict__ x,
                const float* __restrict__ ws,
                float* __restrict__ gate,     // (B, 2T)
                float* __restrict__ ext_out)  // (B, 2T, 5)
{
    __shared__ float sh_in [TILE_DW];
    __shared__ float sh_out[TILE_DW];

    const int tid  = threadIdx.x;
    const int b    = blockIdx.x >> 6;        // 64 tiles per batch row
    const int tile = blockIdx.x & 63;
    const int te0  = tile << 8;              // first te of this tile
    const bool mirror = (te0 >= TLEN);
    const int mode = ((const int*)ws)[10];

    // Contiguous source row range in x for this tile.
    int row0;
    if (!mirror) {
        row0 = te0;                              // first half: x[b, te]
    } else {
        const int t0 = te0 - TLEN;
        row0 = (mode == 1) ? (TLEN - TILE - t0)  // flip: reversed block
                           : t0;                 // 9-x or identity: same rows
    }
    const float* src = x + ((size_t)b * TLEN + (size_t)row0) * 5;

    // ---- async global -> LDS staging: 1280 dwords, 5 per thread ----
    const uint32_t in_base = (uint32_t)(uintptr_t)(&sh_in[0]);  // low 32b = LDS addr
    #pragma unroll
    for (int i = 0; i < 5; ++i) {
        const int chunk = tid + i * TILE;                 // 0..1279
        uint32_t lds_addr = in_base + (uint32_t)(chunk * 4);
        unsigned long long gaddr =
            (unsigned long long)(uintptr_t)src + (unsigned long long)(chunk * 4);
        asm volatile("global_load_async_to_lds_b32 %0, %1, off"
                     :: "v"(lds_addr), "v"(gaddr) : "memory");
    }
    asm volatile("s_wait_asynccnt 0" ::: "memory");
    __syncthreads();

    // ---- fetch this thread's element (reversed slot for flip mode) ----
    const int slot = (mirror && mode == 1) ? (TILE - 1 - tid) : tid;
    float v0 = sh_in[slot * 5 + 0];
    float v1 = sh_in[slot * 5 + 1];
    float v2 = sh_in[slot * 5 + 2];
    float v3 = sh_in[slot * 5 + 3];
    float v4 = sh_in[slot * 5 + 4];
    if (mirror && mode == 0) {               // mirrored = 9 - x
        v0 = 9.0f - v0; v1 = 9.0f - v1; v2 = 9.0f - v2;
        v3 = 9.0f - v3; v4 = 9.0f - v4;
    }

    // extended_x values go to the output staging buffer (in te order).
    sh_out[tid * 5 + 0] = v0;
    sh_out[tid * 5 + 1] = v1;
    sh_out[tid * 5 + 2] = v2;
    sh_out[tid * 5 + 3] = v3;
    sh_out[tid * 5 + 4] = v4;

    // ---- folded weights for this half ----
    const float* w = mirror ? (ws + 5) : ws;
    const float w0 = w[0], w1 = w[1], w2 = w[2], w3 = w[3], w4 = w[4];

    // ---- 10-step scan: r = w.v ; v = [mod(r,10), v0, v1, v2, v3] ----
    float h[N_STEPS];
    #pragma unroll
    for (int k = 0; k < N_STEPS; ++k) {
        float r = fmaf(w0, v0, fmaf(w1, v1, fmaf(w2, v2, fmaf(w3, v3, w4 * v4))));
        h[k] = r;
        float m = mod10(r);
        v4 = v3; v3 = v2; v2 = v1; v1 = v0; v0 = m;
    }

    // ---- score = 1 / (1 + std(h, ddof=1)) ----
    float mean = 0.0f;
    #pragma unroll
    for (int k = 0; k < N_STEPS; ++k) mean += h[k];
    mean *= (1.0f / N_STEPS);
    float var = 0.0f;
    #pragma unroll
    for (int k = 0; k < N_STEPS; ++k) {
        float d = h[k] - mean;
        var += d * d;
    }
    var *= (1.0f / (N_STEPS - 1));
    const float score = 1.0f / (1.0f + sqrtf(var));

    // gate output: one dword per lane, fully coalesced.
    const size_t eidx = (size_t)b * TWO_T + (size_t)te0 + (size_t)tid;
    gate[eidx] = score;

    // ---- async LDS -> global drain of extended_x tile (contiguous dwords) ----
    __syncthreads();   // sh_out fully written by all threads
    const uint32_t out_base = (uint32_t)(uintptr_t)(&sh_out[0]);
    float* dst = ext_out + ((size_t)b * TWO_T + (size_t)te0) * 5;
    #pragma unroll
    for (int i = 0; i < 5; ++i) {
        const int chunk = tid + i * TILE;                 // 0..1279
        uint32_t lds_addr = out_base + (uint32_t)(chunk * 4);
        unsigned long long gaddr =
            (unsigned long long)(uintptr_t)dst + (unsigned long long)(chunk * 4);
        asm volatile("global_store_async_from_lds_b32 %0, %1, off"
                     :: "v"(gaddr), "v"(lds_addr) : "memory");
    }
    // S_ENDPGM performs an implicit wait-idle (ASYNCcnt -> 0) before the
    // wave terminates, so the async drain is guaranteed to complete.
}

extern "C" void kernel_launch(void* const* d_in, const int* in_sizes, int n_in,
                              void* d_out, int out_size, void* d_ws, size_t ws_size,
                              hipStream_t stream) {
    const float* x      = (const float*)d_in[0];
    const float* temple = (const float*)d_in[1];
    const float* breath = (const float*)d_in[2];
    float* out = (float*)d_out;
    float* wsf = (float*)d_ws;

    TempleGate_prep<<<1, 1, 0, stream>>>(x, temple, breath, wsf, out, out_size);

    const size_t gate_elems = (size_t)BATCH * TWO_T;   // 4,194,304
    float* gate = out;
    float* ext  = out + gate_elems;
    const int blocks = BATCH * (TWO_T / TILE);          // 16384
    TempleGate_main<<<blocks, TILE, 0, stream>>>(x, wsf, gate, ext);
}